// NearestUpsampling_31739808317549
// MI455X (gfx1250) — compile-verified
//
#include <hip/hip_runtime.h>

// Segment-mean (scatter_mean) for C=32 channels.
//   out[t, :] = mean over valid edges e with tgt_ids[e]==t of feat[src_ids[e], :]
// Valid edge: all 3 ntypes entries >= 0.
//
// Memory/atomic-bound: ~2-3 GB traffic, no dense FLOPs -> optimal MI455X path is
// 128B-granular gathers + native global_atomic_add_f32, with NT temporal hints
// on the stream-once edge arrays so the 256MB feat table (reused ~4x/row) owns
// the 192MB L2.

#define CHANNELS 32
#define LANES_PER_ROW 8   // 8 lanes x float4 = one 128B feature row

// Native clang vector type: required by __builtin_nontemporal_* and maps
// directly to b128 global memory ops.
typedef float v4f __attribute__((ext_vector_type(4)));

__global__ __launch_bounds__(256) void scatter_edges_kernel(
    const float* __restrict__ feat,        // [N_SRC, 32]
    const int*   __restrict__ src_ids,     // [E]
    const int*   __restrict__ tgt_ids,     // [E]
    const int*   __restrict__ ntypes,      // [E, 3]
    float*       __restrict__ sums,        // [N_TGT, 32]  (pre-zeroed = d_out)
    unsigned int* __restrict__ counts,     // [N_TGT]      (pre-zeroed, in d_ws)
    int num_edges)
{
    const int tid = blockIdx.x * blockDim.x + threadIdx.x;
    const int e   = tid >> 3;          // edge index (8 lanes per edge)
    const int sub = tid & 7;           // float4 slot within the 32-ch row
    if (e >= num_edges) return;

    // Validity mask: valid iff all 3 entries >= 0 (OR of sign bits).
    // Stream-once data: non-temporal loads keep these lines out of L2 so the
    // feat gather table stays resident.
    const int* __restrict__ nt = ntypes + 3LL * e;
    const int t0 = __builtin_nontemporal_load(nt + 0);
    const int t1 = __builtin_nontemporal_load(nt + 1);
    const int t2 = __builtin_nontemporal_load(nt + 2);
    if ((t0 | t1 | t2) < 0) return;

    const int s = __builtin_nontemporal_load(src_ids + e);
    const int t = __builtin_nontemporal_load(tgt_ids + e);

    // Gather one 128B row: 8 adjacent lanes issue contiguous b128 loads.
    // Default (RT) temporal hint: feat rows are reused ~4x -> want L2 residency.
    const v4f v = ((const v4f*)(feat + (size_t)s * CHANNELS))[sub];

    // Scatter-add into the target row: contiguous 128B of f32 atomics
    // (no-return -> STOREcnt path, RMW serviced at L2).
    float* __restrict__ dst = sums + (size_t)t * CHANNELS + sub * 4;
    atomicAdd(dst + 0, v.x);
    atomicAdd(dst + 1, v.y);
    atomicAdd(dst + 2, v.z);
    atomicAdd(dst + 3, v.w);

    if (sub == 0) {
        atomicAdd(counts + t, 1u);
    }
}

__global__ __launch_bounds__(256) void finalize_mean_kernel(
    float*              __restrict__ out,     // [N_TGT, 32] in-place sums -> means
    const unsigned int* __restrict__ counts,  // [N_TGT]
    int n_tgt)
{
    const int tid = blockIdx.x * blockDim.x + threadIdx.x;
    const int i   = tid >> 3;
    const int sub = tid & 7;
    if (i >= n_tgt) return;

    const unsigned int c = counts[i];
    const float inv = (c > 0u) ? (1.0f / (float)c) : 0.0f;  // empty segment -> 0

    v4f* __restrict__ p = ((v4f*)(out + (size_t)i * CHANNELS)) + sub;
    v4f v = *p;                    // dirty in L2 from the atomic pass -> hits
    v *= inv;
    // Output is written once and never re-read on device: NT store avoids
    // holding dirty lines in L2.
    __builtin_nontemporal_store(v, p);
}

extern "C" void kernel_launch(void* const* d_in, const int* in_sizes, int n_in,
                              void* d_out, int out_size, void* d_ws, size_t ws_size,
                              hipStream_t stream) {
    // Inputs in setup_inputs() order:
    //   0: feat    float32 [N_SRC*32]
    //   1: src_ids int32   [E]
    //   2: tgt_ids int32   [E]
    //   3: ntypes  int32   [E*3]
    //   4: n_tgt   int32   [1]   (derive from out_size instead; avoids a D2H read)
    const float* feat    = (const float*)d_in[0];
    const int*   src_ids = (const int*)d_in[1];
    const int*   tgt_ids = (const int*)d_in[2];
    const int*   ntypes  = (const int*)d_in[3];

    const int E     = in_sizes[1];
    const int n_tgt = out_size / CHANNELS;

    float*        out    = (float*)d_out;
    unsigned int* counts = (unsigned int*)d_ws;   // n_tgt * 4 bytes of scratch

    // Zero accumulators every call (graph-capture-safe memset nodes).
    (void)hipMemsetAsync(out, 0, (size_t)out_size * sizeof(float), stream);
    (void)hipMemsetAsync(counts, 0, (size_t)n_tgt * sizeof(unsigned int), stream);

    // Edge scatter: 8 lanes per edge.
    {
        const long long total = (long long)E * LANES_PER_ROW;
        const int threads = 256;
        const int blocks  = (int)((total + threads - 1) / threads);
        scatter_edges_kernel<<<blocks, threads, 0, stream>>>(
            feat, src_ids, tgt_ids, ntypes, out, counts, E);
    }

    // Finalize: divide each non-empty segment row by its count.
    {
        const long long total = (long long)n_tgt * LANES_PER_ROW;
        const int threads = 256;
        const int blocks  = (int)((total + threads - 1) / threads);
        finalize_mean_kernel<<<blocks, threads, 0, stream>>>(out, counts, n_tgt);
    }
}